// SelfNormAttention2_10505490006115
// MI455X (gfx1250) — compile-verified
//
#include <hip/hip_runtime.h>
#include <math.h>

// D = A(16x4,f32) * B(4x16,f32) + C(16x16,f32)
typedef __attribute__((ext_vector_type(2))) float v2f;
typedef __attribute__((ext_vector_type(8))) float v8f;
typedef __attribute__((ext_vector_type(4))) float f4;
typedef __attribute__((ext_vector_type(4))) int   v4i;

#define WMMA_F32(a, b, c) \
  __builtin_amdgcn_wmma_f32_16x16x4_f32(false, (a), false, (b), (short)0, (c), false, false)

constexpr int B_ = 2, S_ = 2048, H_ = 8, D_ = 64;
constexpr int C_ = 64;            // chunk length
constexpr int NC_ = S_ / C_;      // 32 chunks
constexpr int P_ = 68;            // LDS pitch: 272B rows (16B aligned), stride-4 bank step
                                  // -> conflict-free column (A-fragment) access

// ---- 16-byte global->LDS copy: async on gfx1250, float4 fallback elsewhere ----
#if __has_builtin(__builtin_amdgcn_global_load_async_to_lds_b128)
#define HAVE_ASYNC 1
typedef __attribute__((address_space(1))) v4i gv4i_t;
typedef __attribute__((address_space(3))) v4i sv4i_t;
__device__ __forceinline__ void copy16(const float* g, float* l) {
  __builtin_amdgcn_global_load_async_to_lds_b128(
      (gv4i_t*)(uintptr_t)g,
      (sv4i_t*)(unsigned)(uintptr_t)l,   // low 32 bits of flat addr = LDS offset
      0, 0);
}
#else
#define HAVE_ASYNC 0
__device__ __forceinline__ void copy16(const float* g, float* l) {
  *(f4*)l = *(const f4*)g;
}
#endif

__device__ __forceinline__ void copy_wait() {
#if HAVE_ASYNC
#if __has_builtin(__builtin_amdgcn_s_wait_asynccnt)
  __builtin_amdgcn_s_wait_asynccnt(0);
#else
  asm volatile("s_wait_asynccnt 0x0" ::: "memory");
#endif
#endif
}

// ---------------------------------------------------------------------------
// Kernel 1: per-chunk KV = K^T V  (D x D), one block per (b,h,chunk)
// ---------------------------------------------------------------------------
__global__ __launch_bounds__(256) void kv_chunk_kernel(const float* __restrict__ qk,
                                                       const float* __restrict__ vin,
                                                       float* __restrict__ kvws) {
  __shared__ __align__(16) float sK[C_ * P_];
  __shared__ __align__(16) float sV[C_ * P_];
  const int c  = blockIdx.x % NC_;
  const int bh = blockIdx.x / NC_;
  const int b  = bh / H_;
  const int h  = bh % H_;
  const int s0 = c * C_;
  const int tid = threadIdx.x;

  // 128-bit async copies: 1024 x 16B per matrix, coalesced
#pragma unroll
  for (int j = 0; j < 4; ++j) {
    int idx = j * 256 + tid;
    int r = idx >> 4, col = (idx & 15) * 4;
    copy16(&qk[((((size_t)b * S_ + s0 + r) * 2 + 1) * H_ + h) * D_ + col], &sK[r * P_ + col]);
    copy16(&vin[(((size_t)b * S_ + s0 + r) * H_ + h) * D_ + col], &sV[r * P_ + col]);
  }
  copy_wait();
  __syncthreads();

  const int wave = tid >> 5;
  const int lane = tid & 31;
  const int lm = lane & 15;
  const int kh = lane >> 4;
  float* dst = kvws + ((size_t)bh * NC_ + c) * (D_ * D_);

  // KV[d][t] = sum_s K[s][d] * V[s][t]; 16 tiles, 8 waves, 2 tiles each
  for (int t = wave; t < 16; t += 8) {
    const int m0 = (t >> 2) * 16;  // d
    const int n0 = (t & 3) * 16;   // t
    v8f acc = {};
    for (int k0 = 0; k0 < C_; k0 += 4) {
      v2f a, bb;
      a.x  = sK[(k0 + 2 * kh + 0) * P_ + (m0 + lm)];
      a.y  = sK[(k0 + 2 * kh + 1) * P_ + (m0 + lm)];
      bb.x = sV[(k0 + 2 * kh + 0) * P_ + (n0 + lm)];
      bb.y = sV[(k0 + 2 * kh + 1) * P_ + (n0 + lm)];
      acc = WMMA_F32(a, bb, acc);
    }
#pragma unroll
    for (int r = 0; r < 8; ++r) {
      int row = m0 + r + 8 * kh;
      dst[row * D_ + n0 + lm] = acc[r];
    }
  }
}

// ---------------------------------------------------------------------------
// Kernel 2: exclusive prefix-sum of chunk states, one block per (b,h), float4
// ---------------------------------------------------------------------------
__global__ __launch_bounds__(256) void scan_kernel(const float* __restrict__ kvws,
                                                   float* __restrict__ pref) {
  const int bh = blockIdx.x;
  const int tid = threadIdx.x;
  f4 run[4];
#pragma unroll
  for (int j = 0; j < 4; ++j) run[j] = (f4){0.f, 0.f, 0.f, 0.f};
  const size_t base4 = (size_t)bh * NC_ * (D_ * D_ / 4);
  const f4* src = (const f4*)kvws;
  f4* dst = (f4*)pref;
  for (int c = 0; c < NC_; ++c) {
    size_t cb = base4 + (size_t)c * (D_ * D_ / 4);
#pragma unroll
    for (int j = 0; j < 4; ++j) {
      size_t e = cb + j * 256 + tid;
      dst[e] = run[j];
      run[j] += src[e];
    }
  }
}

// ---------------------------------------------------------------------------
// Kernel 3: out = exp(-n) * ( Q * S_prefix + causal_mask(Q K^T) * V )
// ---------------------------------------------------------------------------
__global__ __launch_bounds__(256) void out_kernel(const float* __restrict__ qk,
                                                  const float* __restrict__ vin,
                                                  const float* __restrict__ nvec,
                                                  const float* __restrict__ pref,
                                                  float* __restrict__ out) {
  __shared__ __align__(16) float sQ[C_ * P_];
  __shared__ __align__(16) float sK[C_ * P_];
  __shared__ __align__(16) float sV[C_ * P_];
  __shared__ __align__(16) float sS[C_ * P_];  // prefix state, then masked A = QK^T
  const int c  = blockIdx.x % NC_;
  const int bh = blockIdx.x / NC_;
  const int b  = bh / H_;
  const int h  = bh % H_;
  const int s0 = c * C_;
  const int tid = threadIdx.x;
  const float* sp = pref + ((size_t)bh * NC_ + c) * (D_ * D_);

#pragma unroll
  for (int j = 0; j < 4; ++j) {
    int idx = j * 256 + tid;
    int r = idx >> 4, col = (idx & 15) * 4;
    copy16(&qk[((((size_t)b * S_ + s0 + r) * 2 + 0) * H_ + h) * D_ + col], &sQ[r * P_ + col]);
    copy16(&qk[((((size_t)b * S_ + s0 + r) * 2 + 1) * H_ + h) * D_ + col], &sK[r * P_ + col]);
    copy16(&vin[(((size_t)b * S_ + s0 + r) * H_ + h) * D_ + col], &sV[r * P_ + col]);
    copy16(&sp[idx * 4], &sS[r * P_ + col]);
  }
  copy_wait();
  __syncthreads();

  const int wave = tid >> 5, lane = tid & 31;
  const int lm = lane & 15, kh = lane >> 4;
  v8f acc[2];   // output tiles (start with inter-chunk term)
  v8f amat[2];  // unmasked QK^T tiles

  // Pass 1: acc = Q * S_prefix  and  amat = Q * K^T   (shared A-fragment = Q)
  for (int ti = 0; ti < 2; ++ti) {
    int t = wave + ti * 8;
    int m0 = (t >> 2) * 16, n0 = (t & 3) * 16;
    v8f a0 = {};
    v8f a1 = {};
    for (int k0 = 0; k0 < D_; k0 += 4) {
      v2f af, bf, bk;
      af.x = sQ[(m0 + lm) * P_ + k0 + 2 * kh + 0];
      af.y = sQ[(m0 + lm) * P_ + k0 + 2 * kh + 1];
      bf.x = sS[(k0 + 2 * kh + 0) * P_ + n0 + lm];       // B[d][t] = S[d][t]
      bf.y = sS[(k0 + 2 * kh + 1) * P_ + n0 + lm];
      a0 = WMMA_F32(af, bf, a0);
      bk.x = sK[(n0 + lm) * P_ + k0 + 2 * kh + 0];       // B[d][j] = K[j][d]
      bk.y = sK[(n0 + lm) * P_ + k0 + 2 * kh + 1];
      a1 = WMMA_F32(af, bk, a1);
    }
    acc[ti] = a0;
    amat[ti] = a1;
  }
  __syncthreads();  // everyone done reading sS (prefix state)

  // Spill QK^T through LDS with causal mask (j <= i); converts C/D->A layout
  for (int ti = 0; ti < 2; ++ti) {
    int t = wave + ti * 8;
    int m0 = (t >> 2) * 16, n0 = (t & 3) * 16;
#pragma unroll
    for (int r = 0; r < 8; ++r) {
      int row = m0 + r + 8 * kh, col = n0 + lm;
      sS[row * P_ + col] = (col <= row) ? amat[ti][r] : 0.f;
    }
  }
  __syncthreads();

  // Pass 2: acc += mask(QK^T) * V ; skip all-zero upper-triangular k-blocks
  for (int ti = 0; ti < 2; ++ti) {
    int t = wave + ti * 8;
    int m0 = (t >> 2) * 16, n0 = (t & 3) * 16;
    v8f a0 = acc[ti];
    const int kmax = m0 + 16;  // A[i][k] == 0 for k > i, rows in [m0, m0+15]
    for (int k0 = 0; k0 < kmax; k0 += 4) {
      v2f af, bf;
      af.x = sS[(m0 + lm) * P_ + k0 + 2 * kh + 0];
      af.y = sS[(m0 + lm) * P_ + k0 + 2 * kh + 1];
      bf.x = sV[(k0 + 2 * kh + 0) * P_ + n0 + lm];
      bf.y = sV[(k0 + 2 * kh + 1) * P_ + n0 + lm];
      a0 = WMMA_F32(af, bf, a0);
    }
    // Epilogue: scale by exp(-n) and store (coalesced across lanes)
#pragma unroll
    for (int r = 0; r < 8; ++r) {
      int row = m0 + r + 8 * kh, col = n0 + lm;
      int s = s0 + row;
      float nv = nvec[((size_t)b * S_ + s) * H_ + h];
      out[(((size_t)b * S_ + s) * H_ + h) * D_ + col] = __expf(-nv) * a0[r];
    }
  }
}

// ---------------------------------------------------------------------------
extern "C" void kernel_launch(void* const* d_in, const int* in_sizes, int n_in,
                              void* d_out, int out_size, void* d_ws, size_t ws_size,
                              hipStream_t stream) {
  const float* qk = (const float*)d_in[0];
  const float* v  = (const float*)d_in[1];
  const float* nn = (const float*)d_in[2];
  float* out = (float*)d_out;

  float* kvws = (float*)d_ws;                                  // 8 MB chunk states
  float* pref = kvws + (size_t)B_ * H_ * NC_ * D_ * D_;        // 8 MB exclusive prefixes

  dim3 blk(256);
  kv_chunk_kernel<<<dim3(B_ * H_ * NC_), blk, 0, stream>>>(qk, v, kvws);
  scan_kernel<<<dim3(B_ * H_), blk, 0, stream>>>(kvws, pref);
  out_kernel<<<dim3(B_ * H_ * NC_), blk, 0, stream>>>(qk, v, nn, pref, out);
}